// CliffordMessagePassing_31104153157801
// MI455X (gfx1250) — compile-verified
//
#include <hip/hip_runtime.h>
#include <hip/hip_bf16.h>

typedef __attribute__((ext_vector_type(2))) float v2f;
typedef __attribute__((ext_vector_type(8))) float v8f;

#define EMB 256
#define CCH 64
#define XPAD 4          // LDS row stride 260 -> banks advance by 4 per row

static __device__ inline v8f wmma4(v2f a, v2f b, v8f c) {
    // D(16x16,f32) = A(16x4,f32) * B(4x16,f32) + C
    return __builtin_amdgcn_wmma_f32_16x16x4_f32(false, a, false, b, (short)0, c,
                                                 false, false);
}

// ---------------------------------------------------------------------------
// Build the Clifford-structured 256x256 kernel matrix directly in
// WMMA-B-fragment-packed order:
//   Bpack[((k4*256 + out_f)*2 + half)*2 + j] = K^T[in_f][out_f]
//   with in_f = 4*k4 + 2*half + j
// so lane (col, half) of a wave loads its (b.x,b.y) pair as ONE b64, and a
// wave's 32 lanes cover a contiguous 256B block.
// K^T[in_f][out_f] = sign[bo][bi] * w[idx[bo][bi]][co][ci],
//   out_f = bo*64+co, in_f = bi*64+ci (blade-major), g0=g1=1.
// ---------------------------------------------------------------------------
__global__ void build_kmat_kernel(const float* __restrict__ w, float* __restrict__ Bp) {
    int t = blockIdx.x * blockDim.x + threadIdx.x;   // 65536 threads
    int j    = t & 1;
    int half = (t >> 1) & 1;
    int out_f = (t >> 2) & (EMB - 1);
    int k4   = t >> 10;
    int in_f = k4 * 4 + half * 2 + j;
    int bo = out_f >> 6, co = out_f & (CCH - 1);
    int bi = in_f  >> 6, ci = in_f  & (CCH - 1);
    int ii = (bo << 2) | bi;
    int widx = (0x1B4EB1E4u >> (2 * ii)) & 3;          // idx[bo][bi]
    float sgn = ((0x848u >> ii) & 1u) ? -1.0f : 1.0f;  // sign[bo][bi]
    Bp[t] = sgn * w[(widx * CCH + co) * CCH + ci];
}

// ---------------------------------------------------------------------------
// Degree counts (segment_sum of ones over src / dst).
// ---------------------------------------------------------------------------
__global__ void degree_kernel(const long long* __restrict__ ei, int E,
                              float* __restrict__ degs, float* __restrict__ degd) {
    int e = blockIdx.x * blockDim.x + threadIdx.x;
    if (e < E) {
        atomicAdd(&degs[(int)ei[e]],     1.0f);
        atomicAdd(&degd[(int)ei[E + e]], 1.0f);
    }
}

// ---------------------------------------------------------------------------
// Edge message kernel: 32 edges per block, 256 threads (8 wave32).
// Each wave: 2 M-tiles (rows 0-15 / 16-31) x 2 N-tiles (32 output columns)
// sharing B fragments -> 4 WMMA per K-step against 2 LDS b64 + 2 global b64.
// ---------------------------------------------------------------------------
__global__ __launch_bounds__(256)
void edge_msg_kernel(const float* __restrict__ h,
                     const long long* __restrict__ ei,
                     const long long* __restrict__ et,
                     const float* __restrict__ rel,
                     const float* __restrict__ Bpm,     // packed msg kernel
                     const float* __restrict__ msg_b,
                     const float* __restrict__ gate_w,
                     const float* __restrict__ gate_b,
                     const float* __restrict__ degs,
                     const float* __restrict__ degd,
                     float* __restrict__ agg,
                     int E) {
    __shared__ float xs[32][EMB + XPAD];   // ~33 KB
    __shared__ float gred[32][8];
    __shared__ float sc[32];
    __shared__ int   dn[32];

    const int tid  = threadIdx.x;
    const int medg = tid >> 3;        // edge slot 0..31 (8 threads/edge)
    const int sub  = tid & 7;

    int e  = blockIdx.x * 32 + medg;
    int ec = (e < E) ? e : (E - 1);   // clamp (E % 32 == 0 here)
    int src = (int)ei[ec];
    int rt  = (int)et[ec];

    const float* hp = h   + (size_t)src * EMB;
    const float* rp = rel + (size_t)rt  * EMB;

    float gpart = 0.0f;
    #pragma unroll
    for (int i = 0; i < 32; ++i) {
        int f = sub * 32 + i;
        float rv = rp[f];
        xs[medg][f] = hp[f] * rv;
        int blade = f >> 6, c = f & (CCH - 1);
        gpart += gate_w[(c << 2) | blade] * rv;   // gate flatten is (c,blade)
    }
    gred[medg][sub] = gpart;
    if (sub == 0) dn[medg] = (int)ei[E + ec];
    __syncthreads();

    if (tid < 32) {
        float g = 0.0f;
        #pragma unroll
        for (int i = 0; i < 8; ++i) g += gred[tid][i];
        g += gate_b[0];
        g = 1.0f / (1.0f + __expf(-g));
        int e2  = blockIdx.x * 32 + tid;
        int e2c = (e2 < E) ? e2 : (E - 1);
        float dd = degs[(int)ei[e2c]] * degd[(int)ei[E + e2c]];
        dd = dd < 1.0f ? 1.0f : dd;
        sc[tid] = g * rsqrtf(dd) * ((e2 < E) ? 1.0f : 0.0f);
    }
    __syncthreads();

    // ---- GEMM: 32 edges x 256 outputs, K=256 ----
    const int lane = tid & 31;
    const int wave = tid >> 5;        // owns 32 output columns
    const int col  = lane & 15;
    const int half = lane >> 4;       // 0|1
    const int krow = half * 2;        // A/B K rows: 0,1 | 2,3
    const int arow = lane & 15;
    const int n0   = wave * 32;

    v8f a00 = {}, a01 = {}, a10 = {}, a11 = {};
    #pragma unroll 4
    for (int k = 0; k < EMB; k += 4) {
        v2f av0 = *(const v2f*)&xs[arow     ][k + krow];
        v2f av1 = *(const v2f*)&xs[arow + 16][k + krow];
        const v2f* bp = (const v2f*)(Bpm + (size_t)(k >> 2) * 1024 + half * 2);
        v2f b0 = bp[(n0 + col) * 2];
        v2f b1 = bp[(n0 + 16 + col) * 2];
        a00 = wmma4(av0, b0, a00);
        a01 = wmma4(av0, b1, a01);
        a10 = wmma4(av1, b0, a10);
        a11 = wmma4(av1, b1, a11);
    }

    const float bias0 = msg_b[n0 + col];
    const float bias1 = msg_b[n0 + 16 + col];
    #pragma unroll
    for (int v = 0; v < 8; ++v) {
        int r0 = half * 8 + v;        // edge slot, M-tile 0
        int r1 = 16 + r0;             // edge slot, M-tile 1
        float s0 = sc[r0], s1 = sc[r1];
        size_t base0 = (size_t)dn[r0] * EMB;
        size_t base1 = (size_t)dn[r1] * EMB;
        atomicAdd(&agg[base0 + n0 + col],      (a00[v] + bias0) * s0);
        atomicAdd(&agg[base0 + n0 + 16 + col], (a01[v] + bias1) * s0);
        atomicAdd(&agg[base1 + n0 + col],      (a10[v] + bias0) * s1);
        atomicAdd(&agg[base1 + n0 + 16 + col], (a11[v] + bias1) * s1);
    }
}

// ---------------------------------------------------------------------------
// Node update kernel: 32 nodes per block. Reads agg in-place from `out`
// (all reads complete before the barrier; each node owned by one block),
// computes Kt_upd^T @ (h + agg) + b, stores with (c*4+blade) permutation.
// ---------------------------------------------------------------------------
__global__ __launch_bounds__(256)
void node_update_kernel(const float* __restrict__ h,
                        const float* __restrict__ Bpu,   // packed upd kernel
                        const float* __restrict__ upd_b,
                        float* __restrict__ out,
                        int N) {
    __shared__ float xs[32][EMB + XPAD];

    const int tid  = threadIdx.x;
    const int mnod = tid >> 3;
    const int sub  = tid & 7;

    int n  = blockIdx.x * 32 + mnod;
    int nc = (n < N) ? n : (N - 1);
    const float* hp = h   + (size_t)nc * EMB;
    const float* ap = out + (size_t)nc * EMB;
    #pragma unroll
    for (int i = 0; i < 32; ++i) {
        int f = sub * 32 + i;
        xs[mnod][f] = hp[f] + ap[f];
    }
    __syncthreads();

    const int lane = tid & 31;
    const int wave = tid >> 5;
    const int col  = lane & 15;
    const int half = lane >> 4;
    const int krow = half * 2;
    const int arow = lane & 15;
    const int n0   = wave * 32;

    v8f a00 = {}, a01 = {}, a10 = {}, a11 = {};
    #pragma unroll 4
    for (int k = 0; k < EMB; k += 4) {
        v2f av0 = *(const v2f*)&xs[arow     ][k + krow];
        v2f av1 = *(const v2f*)&xs[arow + 16][k + krow];
        const v2f* bp = (const v2f*)(Bpu + (size_t)(k >> 2) * 1024 + half * 2);
        v2f b0 = bp[(n0 + col) * 2];
        v2f b1 = bp[(n0 + 16 + col) * 2];
        a00 = wmma4(av0, b0, a00);
        a01 = wmma4(av0, b1, a01);
        a10 = wmma4(av1, b0, a10);
        a11 = wmma4(av1, b1, a11);
    }

    const int f0 = n0 + col;
    const int f1 = n0 + 16 + col;
    const int p0 = ((f0 & (CCH - 1)) << 2) | (f0 >> 6);  // -> c*4 + blade
    const int p1 = ((f1 & (CCH - 1)) << 2) | (f1 >> 6);
    const float bias0 = upd_b[f0];
    const float bias1 = upd_b[f1];
    #pragma unroll
    for (int v = 0; v < 8; ++v) {
        int r0 = half * 8 + v;
        int r1 = 16 + r0;
        int n0d = blockIdx.x * 32 + r0;  n0d = (n0d < N) ? n0d : (N - 1);
        int n1d = blockIdx.x * 32 + r1;  n1d = (n1d < N) ? n1d : (N - 1);
        out[(size_t)n0d * EMB + p0] = a00[v] + bias0;
        out[(size_t)n0d * EMB + p1] = a01[v] + bias1;
        out[(size_t)n1d * EMB + p0] = a10[v] + bias0;
        out[(size_t)n1d * EMB + p1] = a11[v] + bias1;
    }
}

// ---------------------------------------------------------------------------
extern "C" void kernel_launch(void* const* d_in, const int* in_sizes, int n_in,
                              void* d_out, int out_size, void* d_ws, size_t ws_size,
                              hipStream_t stream) {
    const float*     h      = (const float*)d_in[0];
    const long long* ei     = (const long long*)d_in[1];
    const long long* et     = (const long long*)d_in[2];
    const float*     rel    = (const float*)d_in[3];
    const float*     msg_w  = (const float*)d_in[4];
    const float*     msg_b  = (const float*)d_in[5];
    const float*     upd_w  = (const float*)d_in[6];
    const float*     upd_b  = (const float*)d_in[7];
    const float*     gate_w = (const float*)d_in[8];
    const float*     gate_b = (const float*)d_in[9];

    const int N = in_sizes[0] / EMB;   // 100000
    const int E = in_sizes[2];         // 600000

    float* out  = (float*)d_out;
    float* ws   = (float*)d_ws;
    float* degs = ws;                  // N floats
    float* degd = ws + N;              // N floats
    float* Bpm  = ws + 2 * (size_t)N;  // 256*256 floats (fragment-packed)
    float* Bpu  = Bpm + EMB * EMB;     // 256*256 floats  (~1.3 MB total)

    hipMemsetAsync(out,  0, (size_t)N * EMB * sizeof(float), stream);
    hipMemsetAsync(degs, 0, (size_t)2 * N * sizeof(float), stream);

    build_kmat_kernel<<<EMB * EMB / 256, 256, 0, stream>>>(msg_w, Bpm);
    build_kmat_kernel<<<EMB * EMB / 256, 256, 0, stream>>>(upd_w, Bpu);
    degree_kernel<<<(E + 255) / 256, 256, 0, stream>>>(ei, E, degs, degd);

    edge_msg_kernel<<<(E + 31) / 32, 256, 0, stream>>>(
        h, ei, et, rel, Bpm, msg_b, gate_w, gate_b, degs, degd, out, E);

    node_update_kernel<<<(N + 31) / 32, 256, 0, stream>>>(h, Bpu, upd_b, out, N);
}